// Random_RNN_24455543783767
// MI455X (gfx1250) — compile-verified
//
#include <hip/hip_runtime.h>

// ---- problem dims (match reference) ----
#define N_IN   1024
#define N_ASS  4096
#define N_OUT  512
#define N_SRC  (N_IN + N_ASS)     // 5120
#define N_TOT  (N_SRC + N_OUT)    // 5632
#define BLK    128                // sequential block size
#define NBLK   (N_ASS / BLK)      // 32
#define ROWS_PER_WG 256           // panel row chunk
#define MAX_CHUNKS  ((N_SRC + ROWS_PER_WG - 1) / ROWS_PER_WG)  // 20

typedef __attribute__((ext_vector_type(2))) float v2f;
typedef __attribute__((ext_vector_type(8))) float v8f;

// ---------------------------------------------------------------------------
// init: h[j] = x[j] for inputs, 0 for associative neurons
// ---------------------------------------------------------------------------
__global__ void rnn_init_h(const float* __restrict__ x, float* __restrict__ h) {
    int j = blockIdx.x * 256 + threadIdx.x;
    if (j < N_SRC) h[j] = (j < N_IN) ? x[j] : 0.0f;
}

// ---------------------------------------------------------------------------
// panel: partial[chunk][t] = sum_{j in chunk} relu(W[j, s+t] * h[j])
// for t in 0..127, chunk = blockIdx.x covering rows [base, base+kmax).
// Uses v_wmma_f32_16x16x4_f32 with the broadcast-vector-in-A trick:
//   A[m][k] = v[k] for all m  =>  D[m][n] = sum_k v[k]*B[k][n]  (all rows equal)
// Per-edge relu via relu(w*h) = relu(w)relu(h) + relu(-w)relu(-h).
// 256 threads = 8 waves; wave w owns 16 columns s+16w .. s+16w+15.
// ---------------------------------------------------------------------------
__global__ __launch_bounds__(256) void rnn_panel(const float* __restrict__ W,
                                                 const float* __restrict__ h,
                                                 float* __restrict__ partials,
                                                 int s) {
    const int lane = threadIdx.x & 31;
    const int wave = threadIdx.x >> 5;            // 0..7
    const int col  = s + wave * 16 + (lane & 15); // target column
    const int kk   = (lane >> 4) << 1;            // 0 for lanes 0-15, 2 for 16-31
    const int base = blockIdx.x * ROWS_PER_WG;
    const int rem  = s - base;
    const int kmax = rem < ROWS_PER_WG ? rem : ROWS_PER_WG;   // multiple of 4

    v8f c = {};
    for (int k = 0; k < kmax; k += 4) {
        const int r = base + k + kk;
        const size_t off = (size_t)r * N_TOT + col;
        const float wa = W[off];
        const float wb = W[off + N_TOT];
        const float ha = h[r];
        const float hb = h[r + 1];
        v2f ap = { fmaxf(ha, 0.0f),  fmaxf(hb, 0.0f)  };
        v2f an = { fmaxf(-ha, 0.0f), fmaxf(-hb, 0.0f) };
        v2f bp = { fmaxf(wa, 0.0f),  fmaxf(wb, 0.0f)  };
        v2f bn = { fmaxf(-wa, 0.0f), fmaxf(-wb, 0.0f) };
        c = __builtin_amdgcn_wmma_f32_16x16x4_f32(false, ap, false, bp,
                                                  (short)0, c, false, false);
        c = __builtin_amdgcn_wmma_f32_16x16x4_f32(false, an, false, bn,
                                                  (short)0, c, false, false);
    }
    // D row 0 lives in VGPR0 (c[0]) of lanes 0..15; all rows identical.
    if (lane < 16)
        partials[blockIdx.x * BLK + wave * 16 + lane] = c[0];
}

// ---------------------------------------------------------------------------
// resolve: one workgroup (128 threads) finishes block [s, s+128) sequentially.
// h_i = (sum of panel partials for col i) + sum_{s<=j<i} relu(W[j,i]*h_j)
// Diagonal 128x128 tile staged in 64KB LDS; one barrier per neuron.
// ---------------------------------------------------------------------------
__global__ __launch_bounds__(128) void rnn_resolve(const float* __restrict__ W,
                                                   float* __restrict__ h,
                                                   const float* __restrict__ partials,
                                                   int s, int nchunks) {
    __shared__ float tile[BLK * BLK];   // 64 KB
    __shared__ float hsh[BLK];
    const int t = threadIdx.x;

    // stage diagonal tile (coalesced per row)
    for (int i = 0; i < BLK; ++i)
        tile[i * BLK + t] = W[(size_t)(s + i) * N_TOT + s + t];

    // deterministic reduction of panel partials for column s+t
    float p = 0.0f;
    for (int cch = 0; cch < nchunks; ++cch)
        p += partials[cch * BLK + t];
    __syncthreads();

    float a = 0.0f;  // intra-block accumulation for column s+t
    for (int i = 0; i < BLK; ++i) {
        if (t == i) hsh[i] = p + a;      // h_{s+i} is now final
        __syncthreads();
        const float hi = hsh[i];
        if (t > i) a += fmaxf(tile[i * BLK + t] * hi, 0.0f);
    }
    h[s + t] = p + a;
}

// ---------------------------------------------------------------------------
// output matvec: pout[chunk][k] = sum_{j in chunk} relu(W[j, 5120+k]*h[j])
// 512 output cols = 32 col-groups; grid = (20 row chunks, 4) x 256 threads.
// ---------------------------------------------------------------------------
__global__ __launch_bounds__(256) void rnn_outmv(const float* __restrict__ W,
                                                 const float* __restrict__ h,
                                                 float* __restrict__ pout) {
    const int lane = threadIdx.x & 31;
    const int wave = threadIdx.x >> 5;
    const int cg   = blockIdx.y * 8 + wave;              // 0..31
    const int col  = N_SRC + cg * 16 + (lane & 15);
    const int kk   = (lane >> 4) << 1;
    const int base = blockIdx.x * ROWS_PER_WG;           // rows, 20 chunks exactly

    v8f c = {};
    for (int k = 0; k < ROWS_PER_WG; k += 4) {
        const int r = base + k + kk;
        const size_t off = (size_t)r * N_TOT + col;
        const float wa = W[off];
        const float wb = W[off + N_TOT];
        const float ha = h[r];
        const float hb = h[r + 1];
        v2f ap = { fmaxf(ha, 0.0f),  fmaxf(hb, 0.0f)  };
        v2f an = { fmaxf(-ha, 0.0f), fmaxf(-hb, 0.0f) };
        v2f bp = { fmaxf(wa, 0.0f),  fmaxf(wb, 0.0f)  };
        v2f bn = { fmaxf(-wa, 0.0f), fmaxf(-wb, 0.0f) };
        c = __builtin_amdgcn_wmma_f32_16x16x4_f32(false, ap, false, bp,
                                                  (short)0, c, false, false);
        c = __builtin_amdgcn_wmma_f32_16x16x4_f32(false, an, false, bn,
                                                  (short)0, c, false, false);
    }
    if (lane < 16)
        pout[blockIdx.x * N_OUT + cg * 16 + lane] = c[0];
}

// ---------------------------------------------------------------------------
// finish: out[k] = relu( sum over 20 row-chunk partials )
// ---------------------------------------------------------------------------
__global__ void rnn_finish(const float* __restrict__ pout, float* __restrict__ out) {
    int k = blockIdx.x * 256 + threadIdx.x;
    if (k < N_OUT) {
        float a = 0.0f;
        for (int cch = 0; cch < MAX_CHUNKS; ++cch)
            a += pout[cch * N_OUT + k];
        out[k] = fmaxf(a, 0.0f);
    }
}

// ---------------------------------------------------------------------------
extern "C" void kernel_launch(void* const* d_in, const int* in_sizes, int n_in,
                              void* d_out, int out_size, void* d_ws, size_t ws_size,
                              hipStream_t stream) {
    const float* x = (const float*)d_in[0];   // [1,1024]
    const float* W = (const float*)d_in[1];   // [5120,5632]
    float* out = (float*)d_out;               // [512]

    float* ws       = (float*)d_ws;
    float* h        = ws;                               // 5120 floats
    float* partials = ws + N_SRC;                       // 20*128 floats
    float* pout     = partials + MAX_CHUNKS * BLK;      // 20*512 floats

    rnn_init_h<<<(N_SRC + 255) / 256, 256, 0, stream>>>(x, h);

    for (int b = 0; b < NBLK; ++b) {
        const int s = N_IN + BLK * b;
        const int nchunks = (s + ROWS_PER_WG - 1) / ROWS_PER_WG;
        rnn_panel<<<nchunks, 256, 0, stream>>>(W, h, partials, s);
        rnn_resolve<<<1, 128, 0, stream>>>(W, h, partials, s, nchunks);
    }

    rnn_outmv<<<dim3(N_SRC / ROWS_PER_WG, 4), 256, 0, stream>>>(W, h, pout);
    rnn_finish<<<(N_OUT + 255) / 256, 256, 0, stream>>>(pout, out);
}